// Linear_diagonal_weight_23605140258925
// MI455X (gfx1250) — compile-verified
//
#include <hip/hip_runtime.h>

// y[b, i] = input[b, i] * weight[i, i] + bias[i]
// B=8192, N=4096, fp32. Pure HBM-streaming kernel: ~256 MiB moved,
// floor ~11.5 us at 23.3 TB/s. Optimized for b128 NT loads/stores,
// gfx1250 prefetch, and per-thread reuse of diag/bias across a row strip.

#define NCOLS 4096
#define NROWS 8192
#define TPB 256                       // 8 x wave32
#define COLS_PER_BLOCK (TPB * 4)      // 1024 columns (float4 per thread)
#define ROWS_PER_BLOCK 32             // row strip amortizing diag/bias loads
#define PREFETCH_ROWS 8               // prefetch distance in rows

typedef __attribute__((ext_vector_type(4))) float v4f;

__global__ __launch_bounds__(TPB) void diag_linear_kernel(
    const float* __restrict__ in,    // [NROWS, NCOLS]
    const float* __restrict__ w,     // [NCOLS, NCOLS] (only diagonal used)
    const float* __restrict__ bias,  // [NCOLS]
    float* __restrict__ out)         // [NROWS, NCOLS]
{
    const int col  = blockIdx.x * COLS_PER_BLOCK + (int)threadIdx.x * 4;
    const int row0 = blockIdx.y * ROWS_PER_BLOCK;

    // Strided diagonal gather: 4 scalar loads per thread, reused across
    // ROWS_PER_BLOCK rows. After the first strip these hit L2.
    v4f d;
    d.x = w[(size_t)(col + 0) * (NCOLS + 1)];
    d.y = w[(size_t)(col + 1) * (NCOLS + 1)];
    d.z = w[(size_t)(col + 2) * (NCOLS + 1)];
    d.w = w[(size_t)(col + 3) * (NCOLS + 1)];

    const v4f b = *reinterpret_cast<const v4f*>(bias + col);

    const float* __restrict__ src = in  + (size_t)row0 * NCOLS + col;
    float*       __restrict__ dst = out + (size_t)row0 * NCOLS + col;

    #pragma unroll 4
    for (int r = 0; r < ROWS_PER_BLOCK; ++r) {
        // gfx1250 prefetch path (global_prefetch_b8): pull the strip a few
        // rows ahead into the hierarchy while current loads are in flight.
        __builtin_prefetch(src + (size_t)PREFETCH_ROWS * NCOLS, 0, 0);

        // Streaming data, zero reuse: non-temporal b128 load/store
        // (global_load_b128 th:TH_LOAD_NT / global_store_b128 th:TH_STORE_NT)
        v4f x = __builtin_nontemporal_load(reinterpret_cast<const v4f*>(src));

        v4f y;
        y.x = __builtin_fmaf(x.x, d.x, b.x);
        y.y = __builtin_fmaf(x.y, d.y, b.y);
        y.z = __builtin_fmaf(x.z, d.z, b.z);
        y.w = __builtin_fmaf(x.w, d.w, b.w);

        __builtin_nontemporal_store(y, reinterpret_cast<v4f*>(dst));

        src += NCOLS;
        dst += NCOLS;
    }
}

extern "C" void kernel_launch(void* const* d_in, const int* in_sizes, int n_in,
                              void* d_out, int out_size, void* d_ws, size_t ws_size,
                              hipStream_t stream) {
    (void)in_sizes; (void)n_in; (void)out_size; (void)d_ws; (void)ws_size;

    const float* input  = (const float*)d_in[0];  // [8192, 4096]
    const float* weight = (const float*)d_in[1];  // [4096, 4096]
    const float* bias   = (const float*)d_in[2];  // [4096]
    float*       out    = (float*)d_out;          // [8192, 4096]

    dim3 grid(NCOLS / COLS_PER_BLOCK, NROWS / ROWS_PER_BLOCK);  // (4, 256)
    dim3 block(TPB);
    diag_linear_kernel<<<grid, block, 0, stream>>>(input, weight, bias, out);
}